// MNN_71511205478969
// MI455X (gfx1250) — compile-verified
//
#include <hip/hip_runtime.h>
#include <hip/hip_bf16.h>
#include <math.h>

// ---------------------------------------------------------------------------
// MI455X plan:
//  - prep:   S[h,w,c] = w_first*m^4 ; T[h,w] = b_first*m^4 + b_mid*(m^3+m^2+m+1)
//  - conv:   fused (affine -> 5x5 conv -> +bias -> relu -> 2x2 maxpool -> branch sum)
//            conv weights staged in LDS; writes F[64][16][48][48] (NCHW-flat)
//  - FC1/FC2: split-K WMMA f32 16x16x4 GEMM (memory-bound: 302MB w_fc stream)
//  - reduce: sum split-K partials + bias + tanh
//  - FC3:    tiny 64x10 dot-product kernel
// ---------------------------------------------------------------------------

typedef __attribute__((ext_vector_type(2))) float v2f;
typedef __attribute__((ext_vector_type(8))) float v8f;

#define Bsz  64
#define Himg 100
#define Wimg 100
#define Cch  16
#define HPp  48
#define WPp  48
#define NFk  (Cch * HPp * WPp)   // 36864
#define N1   2048
#define N2   512
#define NCLS 10

// ---------------- prep: per-pixel effective affine ----------------
__global__ void prep_affine_kernel(const float* __restrict__ w_first,
                                   const float* __restrict__ b_first,
                                   const float* __restrict__ w_mid,
                                   const float* __restrict__ b_mid,
                                   float* __restrict__ S,   // [100*100*16]
                                   float* __restrict__ T)   // [100*100]
{
    int p = blockIdx.x * 256 + threadIdx.x;
    if (p >= Himg * Wimg) return;
    float m  = w_mid[p];           // (H,W,1)
    float m2 = m * m;
    float m4 = m2 * m2;
    float bf = b_first[0], bm = b_mid[0];
    T[p] = bf * m4 + bm * (m2 * m + m2 + m + 1.0f);
    const float* wf = w_first + p * Cch;
    float* sp = S + p * Cch;
#pragma unroll
    for (int c = 0; c < Cch; ++c) sp[c] = wf[c] * m4;
}

// ------- fused conv(5x5) + bias + relu + 2x2 maxpool + branch sum -------
// one thread per (b, ph, pw); loops both branches internally.
__global__ void conv_pool_kernel(const float* __restrict__ x,   // [64][100][100][2]
                                 const float* __restrict__ S,   // [100][100][16]
                                 const float* __restrict__ T,   // [100][100]
                                 const float* __restrict__ cw,  // [16][16][5][5]
                                 const float* __restrict__ cb,  // [16]
                                 float* __restrict__ F)         // [64][16][48][48]
{
    __shared__ float lcw[Cch * Cch * 25];   // 25.6 KB of 320 KB LDS
    for (int i = threadIdx.x; i < Cch * Cch * 25; i += 256) lcw[i] = cw[i];
    __syncthreads();

    int gid = blockIdx.x * 256 + threadIdx.x;     // 64*48*48 = 147456 exactly
    int pw = gid % WPp;
    int t  = gid / WPp;
    int ph = t % HPp;
    int b  = t / HPp;

    float poolsum[Cch];
#pragma unroll
    for (int co = 0; co < Cch; ++co) poolsum[co] = 0.0f;

    for (int br = 0; br < 2; ++br) {
        float pmax[Cch];
#pragma unroll
        for (int co = 0; co < Cch; ++co) pmax[co] = 0.0f;  // relu outputs are >= 0

        for (int sub = 0; sub < 4; ++sub) {
            int oh = ph * 2 + (sub >> 1);
            int ow = pw * 2 + (sub & 1);
            float acc[Cch];
#pragma unroll
            for (int co = 0; co < Cch; ++co) acc[co] = cb[co];

            for (int kh = 0; kh < 5; ++kh) {
                for (int kw = 0; kw < 5; ++kw) {
                    int h = oh + kh, w = ow + kw;
                    float xv = x[((b * Himg + h) * Wimg + w) * 2 + br];
                    float tv = T[h * Wimg + w];
                    const float* Sp = S + (h * Wimg + w) * Cch;
                    float a[Cch];
#pragma unroll
                    for (int ci = 0; ci < Cch; ++ci) a[ci] = fmaf(xv, Sp[ci], tv);
                    const float* wk = lcw + kh * 5 + kw;
#pragma unroll
                    for (int co = 0; co < Cch; ++co) {
                        const float* wr = wk + co * 400;
#pragma unroll
                        for (int ci = 0; ci < Cch; ++ci)
                            acc[co] = fmaf(a[ci], wr[ci * 25], acc[co]);
                    }
                }
            }
#pragma unroll
            for (int co = 0; co < Cch; ++co)
                pmax[co] = fmaxf(pmax[co], fmaxf(acc[co], 0.0f));
        }
#pragma unroll
        for (int co = 0; co < Cch; ++co) poolsum[co] += pmax[co];
    }
#pragma unroll
    for (int co = 0; co < Cch; ++co)
        F[(size_t)b * NFk + co * (HPp * WPp) + ph * WPp + pw] = poolsum[co];
}

// ---------------- split-K WMMA f32 GEMM: part[ks][64][N] = A[64][K] * W[N][K]^T ----
// blockDim = 128 (4 waves); wave -> one 16-wide N tile; 4 accumulator tiles cover M=64.
// A-frag per lane (16x4 f32, 2 VGPR): v.x=A[m][k+koff], v.y=A[m][k+koff+1], koff=2*(lane/16)
// B-frag mirrored; D: vgpr r, lane L -> [M=r+8*(L/16)][N=L%16]
__global__ void gemm_wmma_f32_kernel(const float* __restrict__ A,   // [64][K]
                                     const float* __restrict__ Wm,  // [N][K]
                                     float* __restrict__ part,      // [ksplit][64][N]
                                     int K, int N, int kchunk)
{
    int wave  = threadIdx.x >> 5;
    int lane  = threadIdx.x & 31;
    int n0    = (blockIdx.x * 4 + wave) * 16;
    int ks    = blockIdx.y;
    int kbeg  = ks * kchunk;
    int kend  = kbeg + kchunk;
    int half  = lane >> 4;
    int idx16 = lane & 15;
    int koff  = half * 2;

    const float* Wrow = Wm + (size_t)(n0 + idx16) * K;
    const float* A0 = A + (size_t)(idx16)      * K;
    const float* A1 = A + (size_t)(idx16 + 16) * K;
    const float* A2 = A + (size_t)(idx16 + 32) * K;
    const float* A3 = A + (size_t)(idx16 + 48) * K;

    v8f acc0 = {}, acc1 = {}, acc2 = {}, acc3 = {};

    for (int k = kbeg; k < kend; k += 16) {
        __builtin_prefetch(Wrow + k + 2048, 0, 0);   // global_prefetch ahead on the 302MB stream
#pragma unroll
        for (int kk = 0; kk < 16; kk += 4) {
            int kx = k + kk + koff;
            v2f bfrag = *(const v2f*)(Wrow + kx);
            v2f a0 = *(const v2f*)(A0 + kx);
            v2f a1 = *(const v2f*)(A1 + kx);
            v2f a2 = *(const v2f*)(A2 + kx);
            v2f a3 = *(const v2f*)(A3 + kx);
            acc0 = __builtin_amdgcn_wmma_f32_16x16x4_f32(false, a0, false, bfrag,
                                                         (short)0, acc0, false, false);
            acc1 = __builtin_amdgcn_wmma_f32_16x16x4_f32(false, a1, false, bfrag,
                                                         (short)0, acc1, false, false);
            acc2 = __builtin_amdgcn_wmma_f32_16x16x4_f32(false, a2, false, bfrag,
                                                         (short)0, acc2, false, false);
            acc3 = __builtin_amdgcn_wmma_f32_16x16x4_f32(false, a3, false, bfrag,
                                                         (short)0, acc3, false, false);
        }
    }

    float* outp = part + (size_t)ks * Bsz * N;
    int mb = 8 * half;
#pragma unroll
    for (int r = 0; r < 8; ++r) {
        outp[(size_t)(mb + r)      * N + n0 + idx16] = acc0[r];
        outp[(size_t)(mb + r + 16) * N + n0 + idx16] = acc1[r];
        outp[(size_t)(mb + r + 32) * N + n0 + idx16] = acc2[r];
        outp[(size_t)(mb + r + 48) * N + n0 + idx16] = acc3[r];
    }
}

// ---------------- split-K reduction + bias + tanh ----------------
__global__ void reduce_bias_tanh_kernel(const float* __restrict__ part,
                                        const float* __restrict__ bias,
                                        float* __restrict__ out,
                                        int N, int nsplit)
{
    int t = blockIdx.x * 256 + threadIdx.x;   // exactly 64*N threads
    int n = t % N;
    float s = bias[n];
    for (int i = 0; i < nsplit; ++i) s += part[(size_t)i * Bsz * N + t];
    out[t] = tanhf(s);
}

// ---------------- FC3: [64][512] x [10][512]^T + b ----------------
__global__ void fc3_kernel(const float* __restrict__ Y2,
                           const float* __restrict__ w3,
                           const float* __restrict__ b3,
                           float* __restrict__ out)
{
    int t = blockIdx.x * 64 + threadIdx.x;    // exactly 640
    int m = t / NCLS, c = t % NCLS;
    float s = b3[c];
    const float* y = Y2 + m * N2;
    const float* w = w3 + c * N2;
#pragma unroll 8
    for (int k = 0; k < N2; ++k) s = fmaf(y[k], w[k], s);
    out[t] = s;
}

// ---------------- workspace layout (floats) ----------------
#define WS_S     0u                      // 160000
#define WS_T     160000u                 // 10000
#define WS_F     170000u                 // 64*36864 = 2359296
#define WS_P1    2529296u                // 8*64*2048 = 1048576
#define WS_Y1    3577872u                // 131072
#define WS_P2    3708944u                // 4*64*512 = 131072
#define WS_Y2    3840016u                // 32768
// total ~3.87M floats (~15.5 MB)

extern "C" void kernel_launch(void* const* d_in, const int* in_sizes, int n_in,
                              void* d_out, int out_size, void* d_ws, size_t ws_size,
                              hipStream_t stream) {
    const float* x       = (const float*)d_in[0];
    const float* w_first = (const float*)d_in[1];
    const float* b_first = (const float*)d_in[2];
    const float* w_mid   = (const float*)d_in[3];
    const float* b_mid   = (const float*)d_in[4];
    const float* conv_w  = (const float*)d_in[5];
    const float* conv_b  = (const float*)d_in[6];
    const float* w_fc    = (const float*)d_in[7];
    const float* b_fc    = (const float*)d_in[8];
    const float* w2      = (const float*)d_in[9];
    const float* b2      = (const float*)d_in[10];
    const float* w3      = (const float*)d_in[11];
    const float* b3      = (const float*)d_in[12];
    float* out = (float*)d_out;

    float* wsf = (float*)d_ws;
    float* S  = wsf + WS_S;
    float* T  = wsf + WS_T;
    float* F  = wsf + WS_F;
    float* P1 = wsf + WS_P1;
    float* Y1 = wsf + WS_Y1;
    float* P2 = wsf + WS_P2;
    float* Y2 = wsf + WS_Y2;

    // 1. effective per-pixel affine
    prep_affine_kernel<<<40, 256, 0, stream>>>(w_first, b_first, w_mid, b_mid, S, T);

    // 2. fused affine+conv+relu+pool+branch-sum -> F[64][36864]
    conv_pool_kernel<<<(Bsz * HPp * WPp) / 256, 256, 0, stream>>>(x, S, T, conv_w, conv_b, F);

    // 3. FC1: 64x36864 * (2048x36864)^T, split-K x8
    gemm_wmma_f32_kernel<<<dim3(N1 / 64, 8), 128, 0, stream>>>(F, w_fc, P1, NFk, N1, NFk / 8);
    reduce_bias_tanh_kernel<<<(Bsz * N1) / 256, 256, 0, stream>>>(P1, b_fc, Y1, N1, 8);

    // 4. FC2: 64x2048 * (512x2048)^T, split-K x4
    gemm_wmma_f32_kernel<<<dim3(N2 / 64, 4), 128, 0, stream>>>(Y1, w2, P2, N1, N2, N1 / 4);
    reduce_bias_tanh_kernel<<<(Bsz * N2) / 256, 256, 0, stream>>>(P2, b2, Y2, N2, 4);

    // 5. FC3
    fc3_kernel<<<10, 64, 0, stream>>>(Y2, w3, b3, out);
}